// EMAQuantizeList_23407571763333
// MI455X (gfx1250) — compile-verified
//
#include <hip/hip_runtime.h>

#define DIM   256
#define NEMB  1024
#define NUM   4
#define BATCH 8
#define SEQ   2048
#define TTOT  (BATCH * SEQ)   // 16384

typedef float v2f __attribute__((ext_vector_type(2)));
typedef float v4f __attribute__((ext_vector_type(4)));
typedef float v8f __attribute__((ext_vector_type(8)));
typedef int   v4i __attribute__((vector_size(16)));   // matches async builtin param type

#if defined(__HIP_DEVICE_COMPILE__) && __has_builtin(__builtin_amdgcn_global_load_async_to_lds_b128)
#define HAS_ASYNC_LDS 1
#else
#define HAS_ASYNC_LDS 0
#endif

// 16-byte global -> LDS stage; async (ASYNCcnt, no VGPR roundtrip) when available.
__device__ __forceinline__ void stage16(const float* g, float* l) {
#if HAS_ASYNC_LDS
    __builtin_amdgcn_global_load_async_to_lds_b128(
        (__attribute__((address_space(1))) v4i*)g,
        (__attribute__((address_space(3))) v4i*)l, 0, 0);
#else
    *(v4f*)l = *(const v4f*)g;
#endif
}

__device__ __forceinline__ void stage_wait() {
#if HAS_ASYNC_LDS
#if __has_builtin(__builtin_amdgcn_s_wait_asynccnt)
    __builtin_amdgcn_s_wait_asynccnt(0);
#else
    asm volatile("s_wait_asynccnt 0x0" ::: "memory");
#endif
#endif
}

__device__ __forceinline__ unsigned long long umin64(unsigned long long a, unsigned long long b) {
    return a < b ? a : b;
}

__device__ __forceinline__ unsigned long long shfl_xor_u64(unsigned long long v, int m) {
    unsigned lo = (unsigned)v;
    unsigned hi = (unsigned)(v >> 32);
    lo = __shfl_xor(lo, m, 32);
    hi = __shfl_xor(hi, m, 32);
    return ((unsigned long long)hi << 32) | lo;
}

// ---------------------------------------------------------------------------
// Kernel 1: build TRANSPOSED Toeplitz matrices WmT[m][d][c] (m = (n,i), 16 of
// them).  WmT[d][c] = w_n[i, c-d+127] if 0 <= c-d+127 < 256 else 0.
// N-major layout => a B fragment (K=kb,kb+1 @ col d) is one contiguous b64.
// ---------------------------------------------------------------------------
__global__ __launch_bounds__(256) void build_wm(const float* __restrict__ w0,
                                                const float* __restrict__ w1,
                                                const float* __restrict__ w2,
                                                const float* __restrict__ w3,
                                                float* __restrict__ wm) {
    int idx = blockIdx.x * 256 + threadIdx.x;   // 16 * 256 * 256 total
    int c = idx & 255;                          // K index (contiguous)
    int d = (idx >> 8) & 255;                   // N index (row)
    int m = idx >> 16;                          // 0..15
    int n, i;
    if (m < 1)      { n = 0; i = m;     }
    else if (m < 4) { n = 1; i = m - 1; }
    else if (m < 9) { n = 2; i = m - 4; }
    else            { n = 3; i = m - 9; }
    const float* w = (n == 0) ? w0 : (n == 1) ? w1 : (n == 2) ? w2 : w3;
    int j = c - d + 127;
    float v = (j >= 0 && j < 256) ? w[i * 256 + j] : 0.0f;
    wm[idx] = v;
}

// ---------------------------------------------------------------------------
// Kernel 2a: embT[n][e][d] = embeds[n][d][e]  (N-major codebook panel)
// ---------------------------------------------------------------------------
__global__ __launch_bounds__(256) void transpose_emb(const float* __restrict__ embeds,
                                                     float* __restrict__ embT) {
    int idx = blockIdx.x * 256 + threadIdx.x;   // NUM*DIM*NEMB = 1M
    int e = idx & (NEMB - 1);
    int d = (idx >> 10) & (DIM - 1);
    int n = idx >> 18;
    embT[((size_t)n * NEMB + e) * DIM + d] = embeds[idx];
}

// ---------------------------------------------------------------------------
// Kernel 2b: e2[n][e] = sum_d embeds[n][d][e]^2   (coalesced over e)
// ---------------------------------------------------------------------------
__global__ __launch_bounds__(256) void compute_e2(const float* __restrict__ embeds,
                                                  float* __restrict__ e2) {
    int idx = blockIdx.x * 256 + threadIdx.x;   // NUM * NEMB = 4096
    if (idx >= NUM * NEMB) return;
    int n = idx >> 10;
    int e = idx & (NEMB - 1);
    const float* p = embeds + (size_t)n * DIM * NEMB + e;
    float s = 0.0f;
    for (int d = 0; d < DIM; ++d) {
        float v = p[(size_t)d * NEMB];
        s += v * v;
    }
    e2[idx] = s;
}

// ---------------------------------------------------------------------------
// Kernel 3: conv as sum of shifted-row Toeplitz GEMMs, WMMA f32 16x16x4.
// Block: 32 t-rows x 256 d-cols, 8 waves; wave = 2 col-tiles x 2 row-tiles,
// every B fragment (single b64 from WmT) feeds 2 WMMAs; A tile async in LDS.
// ---------------------------------------------------------------------------
#define CONV_ROWS 38   // 32 rows + halo of 3 on each side

__global__ __launch_bounds__(256) void conv_wmma(const float* __restrict__ x,
                                                 const float* __restrict__ wm,
                                                 const float* __restrict__ cb0,
                                                 const float* __restrict__ cb1,
                                                 const float* __restrict__ cb2,
                                                 const float* __restrict__ cb3,
                                                 float* __restrict__ flats) {
    __shared__ float xs[CONV_ROWS * DIM];   // rows s0-3 .. s0+34, zero padded

    const int n  = blockIdx.y;
    const int t0 = blockIdx.x * 32;
    const int b  = t0 / SEQ;
    const int s0 = t0 % SEQ;

    // stage x tile (16B granules): async to LDS for in-range rows, zeros else
    for (int s4 = threadIdx.x; s4 < CONV_ROWS * (DIM / 4); s4 += 256) {
        int r  = s4 >> 6;          // LDS row 0..37
        int c4 = s4 & 63;          // 16B column group
        int s  = s0 + r - 3;
        if (s >= 0 && s < SEQ) {
            stage16(x + ((size_t)b * SEQ + s) * DIM + c4 * 4, xs + r * DIM + c4 * 4);
        } else {
            v4f z = {0.0f, 0.0f, 0.0f, 0.0f};
            *(v4f*)(xs + r * DIM + c4 * 4) = z;
        }
    }
    stage_wait();
    __syncthreads();

    const int lane = threadIdx.x & 31;
    const int wave = threadIdx.x >> 5;
    const int hi   = lane >> 4;       // 0/1
    const int row  = lane & 15;
    const int d0   = wave * 32;       // this wave's two 16-col tiles: d0, d0+16

    const int ksz  = (n == 0) ? 1 : (n == 1) ? 3 : (n == 2) ? 5 : 7;
    const int moff = (n == 0) ? 0 : (n == 1) ? 1 : (n == 2) ? 4 : 9;
    const int ph   = (ksz - 1) >> 1;

    v8f acc00 = {}, acc01 = {};   // row-tile 0 x {colA, colB}
    v8f acc10 = {}, acc11 = {};   // row-tile 1 x {colA, colB}

    for (int i = 0; i < ksz; ++i) {
        const float* wrow0 = wm + ((size_t)(moff + i) * DIM + d0 + row) * DIM;
        const float* wrow1 = wrow0 + 16 * DIM;
        const float* arow0 = xs + (row + i - ph + 3) * DIM;
        const float* arow1 = arow0 + 16 * DIM;
#pragma unroll 8
        for (int kk = 0; kk < DIM; kk += 4) {
            int kb = kk + 2 * hi;
            v2f a0 = *(const v2f*)(arow0 + kb);
            v2f a1 = *(const v2f*)(arow1 + kb);
            v2f bA = *(const v2f*)(wrow0 + kb);
            v2f bB = *(const v2f*)(wrow1 + kb);
            acc00 = __builtin_amdgcn_wmma_f32_16x16x4_f32(false, a0, false, bA,
                                                          (short)0, acc00, false, false);
            acc10 = __builtin_amdgcn_wmma_f32_16x16x4_f32(false, a1, false, bA,
                                                          (short)0, acc10, false, false);
            acc01 = __builtin_amdgcn_wmma_f32_16x16x4_f32(false, a0, false, bB,
                                                          (short)0, acc01, false, false);
            acc11 = __builtin_amdgcn_wmma_f32_16x16x4_f32(false, a1, false, bB,
                                                          (short)0, acc11, false, false);
        }
    }

    const float* bp = (n == 0) ? cb0 : (n == 1) ? cb1 : (n == 2) ? cb2 : cb3;
    const float bias = bp[0];

#pragma unroll
    for (int v = 0; v < 8; ++v) {
        int r0 = v + 8 * hi;
        int r1 = r0 + 16;
        size_t base0 = ((size_t)n * TTOT + t0 + r0) * DIM;
        size_t base1 = ((size_t)n * TTOT + t0 + r1) * DIM;
        flats[base0 + d0 + row]      = fmaxf(acc00[v] + bias, 0.0f);
        flats[base0 + d0 + 16 + row] = fmaxf(acc01[v] + bias, 0.0f);
        flats[base1 + d0 + row]      = fmaxf(acc10[v] + bias, 0.0f);
        flats[base1 + d0 + 16 + row] = fmaxf(acc11[v] + bias, 0.0f);
    }
}

// ---------------------------------------------------------------------------
// Kernel 4: fused distance GEMM + argmin over 1024 codes.
// argmin_e (x2 - 2 dot + e2) == argmin_e (e2 - 2 dot)  ->  x2 dropped.
// Block: 32 t-rows; wave w covers codes [128w, 128w+128).
// Col-tile loop is INNERMOST: one A pair (2 ds_b64) feeds 16 WMMAs; each B
// fragment is one b64 from N-major embT and feeds 2 WMMAs.
// Packed u64 key (monotonic(score) << 32 | code) -> min == first argmin.
// ---------------------------------------------------------------------------
__global__ __launch_bounds__(256) void vq_argmin(const float* __restrict__ flats,
                                                 const float* __restrict__ embT,
                                                 const float* __restrict__ e2,
                                                 int* __restrict__ ind) {
    __shared__ float as[32 * DIM];                 // 32 KB A tile
    __shared__ unsigned long long red[8][32];

    const int n  = blockIdx.y;
    const int t0 = blockIdx.x * 32;

    const float* fl = flats + ((size_t)n * TTOT + t0) * DIM;
    for (int s4 = threadIdx.x; s4 < 32 * (DIM / 4); s4 += 256)
        stage16(fl + s4 * 4, as + s4 * 4);
    stage_wait();
    __syncthreads();

    const int lane = threadIdx.x & 31;
    const int wave = threadIdx.x >> 5;
    const int hi   = lane >> 4;
    const int row  = lane & 15;

    const float* embn  = embT + (size_t)n * NEMB * DIM;
    const float* e2n   = e2 + n * NEMB;
    const float* arow0 = as + row * DIM;
    const float* arow1 = arow0 + 16 * DIM;
    const float* erow  = embn + (size_t)(wave * 128 + row) * DIM;   // +j*16*DIM per tile

    v8f acc0[8], acc1[8];
#pragma unroll
    for (int j = 0; j < 8; ++j) { acc0[j] = (v8f){}; acc1[j] = (v8f){}; }

#pragma unroll 2
    for (int kk = 0; kk < DIM; kk += 4) {
        int kb = kk + 2 * hi;
        v2f a0 = *(const v2f*)(arow0 + kb);
        v2f a1 = *(const v2f*)(arow1 + kb);
#pragma unroll
        for (int j = 0; j < 8; ++j) {
            v2f bb = *(const v2f*)(erow + (size_t)j * 16 * DIM + kb);
            acc0[j] = __builtin_amdgcn_wmma_f32_16x16x4_f32(false, a0, false, bb,
                                                            (short)0, acc0[j], false, false);
            acc1[j] = __builtin_amdgcn_wmma_f32_16x16x4_f32(false, a1, false, bb,
                                                            (short)0, acc1[j], false, false);
        }
    }

    unsigned long long best0[8], best1[8];
#pragma unroll
    for (int v = 0; v < 8; ++v) { best0[v] = ~0ULL; best1[v] = ~0ULL; }

#pragma unroll
    for (int j = 0; j < 8; ++j) {
        int col = wave * 128 + j * 16 + row;
        float ecol = e2n[col];
#pragma unroll
        for (int v = 0; v < 8; ++v) {
            float s0 = ecol - 2.0f * acc0[j][v];
            float s1 = ecol - 2.0f * acc1[j][v];
            unsigned u0 = __float_as_uint(s0);
            unsigned u1 = __float_as_uint(s1);
            u0 = u0 ^ ((unsigned)((int)u0 >> 31) | 0x80000000u);
            u1 = u1 ^ ((unsigned)((int)u1 >> 31) | 0x80000000u);
            best0[v] = umin64(best0[v], ((unsigned long long)u0 << 32) | (unsigned)col);
            best1[v] = umin64(best1[v], ((unsigned long long)u1 << 32) | (unsigned)col);
        }
    }

    // reduce over the 16 lanes of each half-wave (cols of each row)
    for (int m = 1; m < 16; m <<= 1)
#pragma unroll
        for (int v = 0; v < 8; ++v) {
            best0[v] = umin64(best0[v], shfl_xor_u64(best0[v], m));
            best1[v] = umin64(best1[v], shfl_xor_u64(best1[v], m));
        }

    if (row == 0)
#pragma unroll
        for (int v = 0; v < 8; ++v) {
            red[wave][v + 8 * hi]      = best0[v];
            red[wave][16 + v + 8 * hi] = best1[v];
        }
    __syncthreads();

    if (threadIdx.x < 32) {
        unsigned long long k = red[0][threadIdx.x];
        for (int w = 1; w < 8; ++w) k = umin64(k, red[w][threadIdx.x]);
        ind[(size_t)n * TTOT + t0 + threadIdx.x] = (int)(k & 0xffffffffu);
    }
}

// ---------------------------------------------------------------------------
// Kernel 5: gather codebook rows (coalesced via embT), straight-through value
// f+(q-f), softmax gate, write all four tuple outputs in flat concat order.
// ---------------------------------------------------------------------------
__global__ __launch_bounds__(256) void finalize(const float* __restrict__ flats,
                                                const float* __restrict__ embT,
                                                const int* __restrict__ ind,
                                                const float* __restrict__ gate_w,
                                                const float* __restrict__ gate_b,
                                                float* __restrict__ out) {
    const int t = blockIdx.x;
    const int d = threadIdx.x;

    __shared__ float lds_red[NUM][8];
    __shared__ float gates[NUM];
    __shared__ int   idxs[NUM];

    if (threadIdx.x < NUM) idxs[threadIdx.x] = ind[(size_t)threadIdx.x * TTOT + t];
    __syncthreads();

    float fv[NUM], qv[NUM], zq[NUM];
    for (int n = 0; n < NUM; ++n) {
        int e = idxs[n];
        float q = embT[((size_t)n * NEMB + e) * DIM + d];   // coalesced row gather
        float f = flats[((size_t)n * TTOT + t) * DIM + d];
        fv[n] = f;
        qv[n] = q;
        zq[n] = f + (q - f);   // straight-through forward value, fp-faithful
    }

    // tuple layout: z_q (T*DIM) | argmin (T*NUM, int32) | inputs_cat | quant_cat
    float* z_q_out     = out;
    int*   argmin_out  = (int*)(out + (size_t)TTOT * DIM);
    float* inputs_cat  = out + (size_t)TTOT * DIM + (size_t)TTOT * NUM;
    float* quant_cat   = inputs_cat + (size_t)TTOT * NUM * DIM;

    for (int n = 0; n < NUM; ++n) {
        inputs_cat[(size_t)t * NUM * DIM + n * DIM + d] = fv[n];
        quant_cat [(size_t)t * NUM * DIM + n * DIM + d] = qv[n];
    }
    if (threadIdx.x < NUM) argmin_out[(size_t)t * NUM + threadIdx.x] = idxs[threadIdx.x];

    // gate logits: dot(zq[0], gate_w[n]) over d  (256-wide block reduction)
    const int lane = threadIdx.x & 31;
    const int wave = threadIdx.x >> 5;
    for (int n = 0; n < NUM; ++n) {
        float p = zq[0] * gate_w[n * DIM + d];
        for (int m = 16; m >= 1; m >>= 1) p += __shfl_xor(p, m, 32);
        if (lane == 0) lds_red[n][wave] = p;
    }
    __syncthreads();
    if (threadIdx.x == 0) {
        float lg[NUM];
        float mx = -3.402823466e+38f;
        for (int n = 0; n < NUM; ++n) {
            float s = 0.0f;
            for (int w = 0; w < 8; ++w) s += lds_red[n][w];
            lg[n] = s + gate_b[n];
            mx = fmaxf(mx, lg[n]);
        }
        float den = 0.0f;
        for (int n = 0; n < NUM; ++n) { lg[n] = __expf(lg[n] - mx); den += lg[n]; }
        for (int n = 0; n < NUM; ++n) gates[n] = lg[n] / den;
    }
    __syncthreads();

    float z = 0.0f;
    for (int n = 0; n < NUM; ++n) z += zq[n] * gates[n];
    z_q_out[(size_t)t * DIM + d] = z;
}

// ---------------------------------------------------------------------------
extern "C" void kernel_launch(void* const* d_in, const int* in_sizes, int n_in,
                              void* d_out, int out_size, void* d_ws, size_t ws_size,
                              hipStream_t stream) {
    const float* x      = (const float*)d_in[0];
    const float* cw0    = (const float*)d_in[1];
    const float* cb0    = (const float*)d_in[2];
    const float* cw1    = (const float*)d_in[3];
    const float* cb1    = (const float*)d_in[4];
    const float* cw2    = (const float*)d_in[5];
    const float* cb2    = (const float*)d_in[6];
    const float* cw3    = (const float*)d_in[7];
    const float* cb3    = (const float*)d_in[8];
    const float* embeds = (const float*)d_in[9];
    const float* gate_w = (const float*)d_in[10];
    const float* gate_b = (const float*)d_in[11];
    float* out = (float*)d_out;

    // workspace layout
    char* ws = (char*)d_ws;
    float* flats = (float*)ws;                                           // 64 MB
    float* wm    = (float*)(ws + (size_t)NUM * TTOT * DIM * 4);          //  4 MB
    float* embT  = (float*)((char*)wm + (size_t)16 * DIM * DIM * 4);     //  4 MB
    float* e2    = (float*)((char*)embT + (size_t)NUM * NEMB * DIM * 4); // 16 KB
    int*   ind   = (int*)((char*)e2 + (size_t)NUM * NEMB * 4);           // 256 KB

    build_wm<<<16 * DIM * DIM / 256, 256, 0, stream>>>(cw0, cw1, cw2, cw3, wm);
    transpose_emb<<<NUM * DIM * NEMB / 256, 256, 0, stream>>>(embeds, embT);
    compute_e2<<<(NUM * NEMB + 255) / 256, 256, 0, stream>>>(embeds, e2);
    conv_wmma<<<dim3(TTOT / 32, NUM), 256, 0, stream>>>(x, wm, cb0, cb1, cb2, cb3, flats);
    vq_argmin<<<dim3(TTOT / 32, NUM), 256, 0, stream>>>(flats, embT, e2, ind);
    finalize<<<TTOT, DIM, 0, stream>>>(flats, embT, ind, gate_w, gate_b, out);
}